// gcn_unweighted_33887291965782
// MI455X (gfx1250) — compile-verified
//
#include <hip/hip_runtime.h>
#include <hip/hip_bf16.h>

typedef __attribute__((ext_vector_type(2))) float v2f;
typedef __attribute__((ext_vector_type(8))) float v8f;

#define IN_FEATS 128
#define H_FEATS  64

// ---------------- utility kernels ----------------

__global__ void zero_f4_kernel(float4* p, long n4) {
    long i = (long)blockIdx.x * blockDim.x + threadIdx.x;
    if (i < n4) p[i] = make_float4(0.f, 0.f, 0.f, 0.f);
}

__global__ void degree_kernel(const long long* __restrict__ src,
                              const long long* __restrict__ dst,
                              unsigned* __restrict__ deg_out,
                              unsigned* __restrict__ deg_in, int E) {
    int i = blockIdx.x * blockDim.x + threadIdx.x;
    if (i < E) {
        atomicAdd(&deg_out[(int)src[i]], 1u);
        atomicAdd(&deg_in[(int)dst[i]], 1u);
    }
}

__global__ void rsqrt_kernel(const unsigned* __restrict__ dout,
                             const unsigned* __restrict__ din,
                             float* __restrict__ rs_out,
                             float* __restrict__ rs_in, int n) {
    int i = blockIdx.x * blockDim.x + threadIdx.x;
    if (i < n) {
        unsigned a = dout[i]; if (a < 1u) a = 1u;
        unsigned b = din[i];  if (b < 1u) b = 1u;
        rs_out[i] = rsqrtf((float)a);
        rs_in[i]  = rsqrtf((float)b);
    }
}

// edge scatter: accum[dst] += xin[src], 64 floats per edge, float4 granularity
__global__ void scatter_kernel(const float* __restrict__ xin,
                               const long long* __restrict__ src,
                               const long long* __restrict__ dst,
                               float* __restrict__ accum, int E) {
    long idx = (long)blockIdx.x * blockDim.x + threadIdx.x;
    long e = idx >> 4;
    if (e >= E) return;
    int q = (int)(idx & 15);
    int s = (int)src[e];
    int d = (int)dst[e];
    float4 v = ((const float4*)(xin + (size_t)s * H_FEATS))[q];
    float* base = accum + (size_t)d * H_FEATS + q * 4;
    atomicAdd(base + 0, v.x);
    atomicAdd(base + 1, v.y);
    atomicAdd(base + 2, v.z);
    atomicAdd(base + 3, v.w);
}

// h = relu(rst*rs_in[row] + b1); x2 = h * rs_out[row]   (fused, float4)
__global__ void act_kernel(const float* __restrict__ rst,
                           const float* __restrict__ rs_in,
                           const float* __restrict__ rs_out,
                           const float* __restrict__ b1,
                           float* __restrict__ x2, int N) {
    long idx = (long)blockIdx.x * blockDim.x + threadIdx.x;
    long n16 = (long)N * 16;
    if (idx >= n16) return;
    int row = (int)(idx >> 4);
    int q = (int)(idx & 15);
    float ri = rs_in[row];
    float ro = rs_out[row];
    float4 v = ((const float4*)(rst + (size_t)row * H_FEATS))[q];
    float4 b = ((const float4*)b1)[q];
    v.x = fmaxf(v.x * ri + b.x, 0.f) * ro;
    v.y = fmaxf(v.y * ri + b.y, 0.f) * ro;
    v.z = fmaxf(v.z * ri + b.z, 0.f) * ro;
    v.w = fmaxf(v.w * ri + b.w, 0.f) * ro;
    ((float4*)(x2 + (size_t)row * H_FEATS))[q] = v;
}

// ---------------- WMMA GEMM kernels ----------------
// V_WMMA_F32_16X16X4_F32 per-lane layout (wave32):
//   half = lane>>4, mr = lane&15
//   A vgpr pair  : x[row = m0+mr][k = 4*s + 2*half + {0,1}]
//   B vgpr pair  : W[k = 4*s + 2*half + {0,1}][col = n0+mr]
//   D vgpr r     : out[row = m0 + half*8 + r][col = n0+mr]

// x1 = (feats .* mask .* rs_out[row]) @ W1   (N x 128) @ (128 x 64)
__global__ __launch_bounds__(256)
void gemm1_wmma(const float* __restrict__ feats,
                const float* __restrict__ mask,
                const float* __restrict__ W1,
                const float* __restrict__ rs_out,
                float* __restrict__ x1, int nrows) {
    __shared__ float w[IN_FEATS * H_FEATS];  // 32 KB
    {
        const float4* s4 = (const float4*)W1;
        float4* d4 = (float4*)w;
        for (int i = threadIdx.x; i < IN_FEATS * H_FEATS / 4; i += 256) d4[i] = s4[i];
    }
    __syncthreads();

    int wave = threadIdx.x >> 5;
    int lane = threadIdx.x & 31;
    int mblk = blockIdx.x * 8 + wave;
    if (mblk * 16 + 15 >= nrows) return;
    int half = lane >> 4;
    int mr = lane & 15;
    int row = mblk * 16 + mr;

    float rs = rs_out[row];
    const v2f* f2 = (const v2f*)(feats + (size_t)row * IN_FEATS);
    const v2f* m2 = (const v2f*)(mask + (size_t)row * IN_FEATS);

    v2f A[32];
#pragma unroll
    for (int s = 0; s < 32; s++) {
        v2f a = f2[2 * s + half];
        v2f m = m2[2 * s + half];
        a.x *= m.x * rs;
        a.y *= m.y * rs;
        A[s] = a;
    }

#pragma unroll
    for (int nt = 0; nt < 4; nt++) {
        v8f acc = {0.f, 0.f, 0.f, 0.f, 0.f, 0.f, 0.f, 0.f};
        int col = nt * 16 + mr;
#pragma unroll
        for (int s = 0; s < 32; s++) {
            int k = 4 * s + 2 * half;
            v2f b;
            b.x = w[k * H_FEATS + col];
            b.y = w[(k + 1) * H_FEATS + col];
            acc = __builtin_amdgcn_wmma_f32_16x16x4_f32(
                false, A[s], false, b, (short)0, acc, false, false);
        }
#pragma unroll
        for (int r = 0; r < 8; r++) {
            int orow = mblk * 16 + half * 8 + r;
            x1[(size_t)orow * H_FEATS + col] = acc[r];
        }
    }
}

// out = (agg @ W2) * rs_in[row] + b2   (N x 64) @ (64 x 128)
__global__ __launch_bounds__(256)
void gemm2_wmma(const float* __restrict__ agg,
                const float* __restrict__ W2,
                const float* __restrict__ rs_in,
                const float* __restrict__ b2,
                float* __restrict__ out, int nrows) {
    __shared__ float w[H_FEATS * IN_FEATS];  // 32 KB
    {
        const float4* s4 = (const float4*)W2;
        float4* d4 = (float4*)w;
        for (int i = threadIdx.x; i < H_FEATS * IN_FEATS / 4; i += 256) d4[i] = s4[i];
    }
    __syncthreads();

    int wave = threadIdx.x >> 5;
    int lane = threadIdx.x & 31;
    int mblk = blockIdx.x * 8 + wave;
    if (mblk * 16 + 15 >= nrows) return;
    int half = lane >> 4;
    int mr = lane & 15;
    int row = mblk * 16 + mr;

    const v2f* a2 = (const v2f*)(agg + (size_t)row * H_FEATS);
    v2f A[16];
#pragma unroll
    for (int s = 0; s < 16; s++) A[s] = a2[2 * s + half];

    float rsv[8];
#pragma unroll
    for (int r = 0; r < 8; r++) rsv[r] = rs_in[mblk * 16 + half * 8 + r];

#pragma unroll
    for (int nt = 0; nt < 8; nt++) {
        v8f acc = {0.f, 0.f, 0.f, 0.f, 0.f, 0.f, 0.f, 0.f};
        int col = nt * 16 + mr;
        float bb = b2[col];
#pragma unroll
        for (int s = 0; s < 16; s++) {
            int k = 4 * s + 2 * half;
            v2f b;
            b.x = w[k * IN_FEATS + col];
            b.y = w[(k + 1) * IN_FEATS + col];
            acc = __builtin_amdgcn_wmma_f32_16x16x4_f32(
                false, A[s], false, b, (short)0, acc, false, false);
        }
#pragma unroll
        for (int r = 0; r < 8; r++) {
            int orow = mblk * 16 + half * 8 + r;
            out[(size_t)orow * IN_FEATS + col] = acc[r] * rsv[r] + bb;
        }
    }
}

// ---------------- launcher ----------------

extern "C" void kernel_launch(void* const* d_in, const int* in_sizes, int n_in,
                              void* d_out, int out_size, void* d_ws, size_t ws_size,
                              hipStream_t stream) {
    const float* feats = (const float*)d_in[0];
    const float* mask  = (const float*)d_in[1];
    const float* W1    = (const float*)d_in[2];
    const float* b1    = (const float*)d_in[3];
    const float* W2    = (const float*)d_in[4];
    const float* b2    = (const float*)d_in[5];
    const long long* src = (const long long*)d_in[6];
    const long long* dst = (const long long*)d_in[7];

    const int N = in_sizes[0] / IN_FEATS;
    const int E = in_sizes[6];
    const int Np = (N + 3) & ~3;  // keep 16B alignment between regions

    unsigned* deg_out = (unsigned*)d_ws;
    unsigned* deg_in  = deg_out + Np;
    float* rs_out = (float*)(deg_in + Np);
    float* rs_in  = rs_out + Np;
    float* bufA   = rs_in + Np;                 // x1, then x2
    float* bufB   = bufA + (size_t)N * H_FEATS; // rst1, then agg2

    float* out = (float*)d_out;

    const int T = 256;
    long degElems = (long)Np * 2;               // deg_out + deg_in (contiguous)
    long aggElems = (long)N * H_FEATS;

    // 1) zero degree counters + first aggregation buffer
    zero_f4_kernel<<<(unsigned)((degElems / 4 + T - 1) / T), T, 0, stream>>>(
        (float4*)deg_out, degElems / 4);
    zero_f4_kernel<<<(unsigned)((aggElems / 4 + T - 1) / T), T, 0, stream>>>(
        (float4*)bufB, aggElems / 4);

    // 2) degrees
    degree_kernel<<<(E + T - 1) / T, T, 0, stream>>>(src, dst, deg_out, deg_in, E);

    // 3) rsqrt(max(deg,1))
    rsqrt_kernel<<<(N + T - 1) / T, T, 0, stream>>>(deg_out, deg_in, rs_out, rs_in, N);

    // 4) GEMM1 (WMMA): bufA = (feats.*mask.*rs_out) @ W1
    int mblks = (N + 15) / 16;
    gemm1_wmma<<<(mblks + 7) / 8, T, 0, stream>>>(feats, mask, W1, rs_out, bufA, N);

    // 5) scatter layer 1: bufB[dst] += bufA[src]
    long sthreads = (long)E * 16;
    scatter_kernel<<<(unsigned)((sthreads + T - 1) / T), T, 0, stream>>>(
        bufA, src, dst, bufB, E);

    // 6) fused relu/bias/norms: bufA = relu(bufB*rs_in + b1) * rs_out
    long athreads = (long)N * 16;
    act_kernel<<<(unsigned)((athreads + T - 1) / T), T, 0, stream>>>(
        bufB, rs_in, rs_out, b1, bufA, N);

    // 7) zero second aggregation buffer (reuse bufB)
    zero_f4_kernel<<<(unsigned)((aggElems / 4 + T - 1) / T), T, 0, stream>>>(
        (float4*)bufB, aggElems / 4);

    // 8) scatter layer 2: bufB[dst] += bufA[src]
    scatter_kernel<<<(unsigned)((sthreads + T - 1) / T), T, 0, stream>>>(
        bufA, src, dst, bufB, E);

    // 9) GEMM2 (WMMA): out = (bufB @ W2) * rs_in + b2
    gemm2_wmma<<<(mblks + 7) / 8, T, 0, stream>>>(bufB, W2, rs_in, b2, out, N);
}